// MAPLoss_69174743269616
// MI455X (gfx1250) — compile-verified
//
#include <hip/hip_runtime.h>
#include <math.h>

#define NEG_INF -1.0e30f
#define BB 32
#define TT 2048
#define CC 1024
#define UU 256
#define SS 513   // 2*U+1

typedef float v2f __attribute__((ext_vector_type(2)));
typedef float v8f __attribute__((ext_vector_type(8)));

// ---------------------------------------------------------------------------
// CDNA5 async copy global -> LDS (VGLOBAL encoding, GVS mode):
//   lds[vdst] = mem[saddr + vaddr]   (16B per lane), tracked with ASYNCcnt.
// saddr stays a loop-invariant SGPR pair; ALL varying offsets go through the
// 32-bit vaddr VGPR so no readfirstlane / 64-bit adds appear in hot loops.
// ---------------------------------------------------------------------------
__device__ __forceinline__ void async_ld_b128(unsigned lds_byte_off,
                                              const void* sbase,
                                              unsigned voff) {
  asm volatile("global_load_async_to_lds_b128 %0, %1, %2"
               :: "v"(lds_byte_off), "v"(voff), "s"(sbase)
               : "memory");
}
#define WAIT_ASYNC(N) asm volatile("s_wait_asynccnt " #N ::: "memory")

__device__ __forceinline__ unsigned lds_off_u32(const void* p) {
  // low 32 bits of a generic LDS address == byte offset in wave's LDS
  return (unsigned)(uintptr_t)p;
}

// ===========================================================================
// Kernel A: per-frame logsumexp (denominator), WMMA reduction.
// Block = 128 threads = 4 waves. Each wave owns 16 frames, loops over 16
// class-tiles of 64. exp() on VALU/TRANS, accumulation on the matrix pipe via
// V_WMMA_F32_16X16X4_F32 with B = ones(4x16):
//   D[m,n] += sum_k A[m,k]  ->  every column of row m holds frame m's sum.
// A fragment layout (ISA 7.12.2, 32-bit 16x4): lane = m + 16*(k>=2),
// VGPR k%2  ->  lane L reads frame (L&15), classes 4j + 2*(L>>4) + {0,1}.
// LDS tile row stride = 68 words (64 + 4 pad) -> conflict-free b64 gathers.
// ===========================================================================
__global__ __launch_bounds__(128) void den_lse_kernel(
    const float* __restrict__ lp,
    const int* __restrict__ input_lengths,
    float* __restrict__ frame_lse) {
  __shared__ float lds[2][4][16 * 68];  // [buf][wave][row*68] = 34816 B

  const int tid  = threadIdx.x;
  const int wave = tid >> 5;
  const int lane = tid & 31;
  const int m    = lane & 15;        // frame row inside wave's 16
  const int kp   = (lane >> 4) * 2;  // K-pair select (0 or 2)

  const long frame0 = (long)blockIdx.x * 64 + (long)wave * 16;
  const float* wbase = lp + frame0 * CC;          // uniform saddr, fixed

  const unsigned col  = (unsigned)(lane & 15) * 16u;  // bytes in 256B row chunk
  const unsigned rsel = (unsigned)(lane >> 4);        // 0/1: second row of pair

  // per-lane invariant parts of the addresses
  const unsigned gbase_lane = rsel * (CC * 4u) + col;   // row-in-pair + col
  unsigned lbase_lane[2];
  lbase_lane[0] = lds_off_u32(&lds[0][wave][0]) + rsel * 272u + col;
  lbase_lane[1] = lds_off_u32(&lds[1][wave][0]) + rsel * 272u + col;

  v8f d = {};  // 16x16 f32 accumulator (frame sums)

  // issue one 64-class tile (16 frames x 256B) as 8 coalesced 512B bursts;
  // saddr fixed, tile/row offsets all in the 32-bit vaddr.
  auto issue_tile = [&](int kk, int buf) {
    const unsigned gtile = gbase_lane + (unsigned)kk * 256u;
#pragma unroll
    for (int i = 0; i < 8; ++i) {
      async_ld_b128(lbase_lane[buf] + (unsigned)i * 544u,   // 2 rows * 272B
                    (const void*)wbase,
                    gtile + (unsigned)i * (2u * CC * 4u));  // 2 rows * 4KB
    }
  };

  auto compute_tile = [&](int buf) {
    const float* base = &lds[buf][wave][0];
    v2f bones;
    bones.x = 1.0f;
    bones.y = 1.0f;
#pragma unroll
    for (int j = 0; j < 16; ++j) {
      v2f x = *(const v2f*)(base + m * 68 + j * 4 + kp);
      v2f a;
      a.x = __expf(x.x);
      a.y = __expf(x.y);
      d = __builtin_amdgcn_wmma_f32_16x16x4_f32(
          /*neg_a=*/false, a, /*neg_b=*/false, bones,
          /*c_mod=*/(short)0, d, /*reuse_a=*/false, /*reuse_b=*/false);
    }
  };

  issue_tile(0, 0);
  for (int kk = 0; kk < 16; ++kk) {
    if (kk + 1 < 16) {
      issue_tile(kk + 1, (kk + 1) & 1);
      WAIT_ASYNC(8);  // 8 outstanding = next tile; current tile complete
    } else {
      WAIT_ASYNC(0);
    }
    compute_tile(kk & 1);
  }

  // extract D[m, 0]: VGPR v / lane 0 -> frame v ; VGPR v / lane 16 -> frame v+8
  float sum = 0.0f;
#pragma unroll
  for (int v = 0; v < 8; ++v) {
    float s0 = __shfl(d[v], 0, 32);
    float s1 = __shfl(d[v], 16, 32);
    if (lane == v) sum = s0;
    if (lane == v + 8) sum = s1;
  }
  if (lane < 16) {
    long f = frame0 + lane;
    int b  = (int)(f >> 11);       // T = 2048
    int t  = (int)(f & (TT - 1));
    float lse = __logf(fmaxf(sum, 1e-37f));
    frame_lse[f] = (t < input_lengths[b]) ? lse : 0.0f;
  }
}

// ===========================================================================
// Kernel B: CTC forward (numerator). One block (256 thr, 8 waves) per batch.
// Frames double-buffered in LDS via async loads (1 x b128 per thread = 4KB),
// frame t+1 DMA overlapped with step t compute. alpha ping-pong in LDS with
// 2 NEG_INF guard slots for branchless s-1/s-2 reads.
// ===========================================================================
__global__ __launch_bounds__(256) void ctc_num_kernel(
    const float* __restrict__ lp,
    const int* __restrict__ targets,
    const int* __restrict__ input_lengths,
    const int* __restrict__ target_lengths,
    float* __restrict__ num_out) {
  __shared__ int ext[SS];
  __shared__ unsigned char same[SS];
  __shared__ float alpha[2][SS + 3];  // [buf][2 guards + 513 + pad]
  __shared__ float fbuf[2][CC];

  const int tid = threadIdx.x;
  const int b   = blockIdx.x;
  const int lenB = input_lengths[b];
  const float* lpb = lp + (size_t)b * TT * CC;   // uniform saddr, fixed

  const unsigned col = (unsigned)tid * 16u;
  const unsigned lfb[2] = {lds_off_u32(&fbuf[0][0]) + col,
                           lds_off_u32(&fbuf[1][0]) + col};

  for (int s = tid; s < SS; s += 256)
    ext[s] = (s & 1) ? targets[b * UU + (s >> 1)] : 0;
  if (tid < 2) {
    alpha[0][tid] = NEG_INF;  // guards, written once
    alpha[1][tid] = NEG_INF;
  }
  __syncthreads();
  for (int s = tid; s < SS; s += 256)
    same[s] = (s < 2) ? 1 : (unsigned char)(ext[s] == ext[s - 2]);

  // prologue: frames 0 and 1
  async_ld_b128(lfb[0], (const void*)lpb, col);
  async_ld_b128(lfb[1], (const void*)lpb, col + CC * 4u);
  WAIT_ASYNC(0);
  __syncthreads();

  for (int s = tid; s < SS; s += 256)
    alpha[0][2 + s] = (s < 2) ? fbuf[0][ext[s]] : NEG_INF;

  for (int t = 1; t < TT; ++t) {
    __syncthreads();  // step t-1 done reading fbuf[(t+1)&1] and alpha bufs
    if (t + 1 < TT) {
      async_ld_b128(lfb[(t + 1) & 1], (const void*)lpb,
                    col + (unsigned)(t + 1) * (CC * 4u));
      WAIT_ASYNC(1);  // frame t (issued last iter) complete for this wave
    } else {
      WAIT_ASYNC(0);
    }
    __syncthreads();  // all waves' chunks of frame t visible

    const float* fr   = fbuf[t & 1];
    const float* aOld = &alpha[(t - 1) & 1][2];
    float* aNew       = &alpha[t & 1][2];
    const bool active = (t < lenB);

    for (int s = tid; s < SS; s += 256) {
      float a0 = aOld[s];
      float r;
      if (active) {
        float a1 = aOld[s - 1];
        float a2 = same[s] ? NEG_INF : aOld[s - 2];
        float mx = fmaxf(a0, fmaxf(a1, a2));
        float sm = __expf(a0 - mx) + __expf(a1 - mx) + __expf(a2 - mx);
        r = mx + __logf(sm) + fr[ext[s]];
      } else {
        r = a0;  // freeze past sequence end
      }
      aNew[s] = r;
    }
  }
  __syncthreads();

  if (tid == 0) {
    int L = target_lengths[b];
    const float* aF = &alpha[(TT - 1) & 1][2];
    float x = aF[2 * L], y = aF[2 * L - 1];
    float mx = fmaxf(x, y);
    num_out[b] = mx + __logf(__expf(x - mx) + __expf(y - mx));
  }
}

// ===========================================================================
// Kernel C1: deterministic per-b denominator sum over T (fixed-order tree).
// ===========================================================================
__global__ __launch_bounds__(256) void den_reduce_kernel(
    const float* __restrict__ frame_lse, float* __restrict__ den) {
  __shared__ float red[256];
  const int b = blockIdx.x, tid = threadIdx.x;
  float acc = 0.0f;
  for (int t = tid; t < TT; t += 256) acc += frame_lse[(size_t)b * TT + t];
  red[tid] = acc;
  __syncthreads();
  for (int s = 128; s > 0; s >>= 1) {
    if (tid < s) red[tid] += red[tid + s];
    __syncthreads();
  }
  if (tid == 0) den[b] = red[0];
}

// ===========================================================================
// Kernel C2: loss + valid mask.  out[0]=loss, out[1..32]=mask as 0/1 floats.
// ===========================================================================
__global__ void finalize_kernel(const float* __restrict__ num,
                                const float* __restrict__ den,
                                float* __restrict__ out) {
  __shared__ float tots[BB];
  __shared__ int vals[BB];
  const int tid = threadIdx.x;
  if (tid < BB) {
    float tot = num[tid] - den[tid];
    int fin = (tot == tot) && (tot < 3.4e38f) && (tot > -3.4e38f);
    int v = fin && (tot > NEG_INF * 0.5f);
    tots[tid] = v ? tot : 0.0f;
    vals[tid] = v;
    out[1 + tid] = v ? 1.0f : 0.0f;
  }
  __syncthreads();
  if (tid == 0) {
    float s = 0.0f;
    int n = 0;
    for (int i = 0; i < BB; ++i) { s += tots[i]; n += vals[i]; }
    if (n < 1) n = 1;
    out[0] = -(s / (float)n);
  }
}

// ===========================================================================
extern "C" void kernel_launch(void* const* d_in, const int* in_sizes, int n_in,
                              void* d_out, int out_size, void* d_ws,
                              size_t ws_size, hipStream_t stream) {
  const float* lp       = (const float*)d_in[0];
  const int* targets    = (const int*)d_in[1];
  const int* in_len     = (const int*)d_in[2];
  const int* tg_len     = (const int*)d_in[3];
  float* out = (float*)d_out;

  float* frame_lse = (float*)d_ws;          // B*T floats
  float* num       = frame_lse + BB * TT;   // B floats
  float* den       = num + BB;              // B floats

  den_lse_kernel<<<(BB * TT) / 64, 128, 0, stream>>>(lp, in_len, frame_lse);
  ctc_num_kernel<<<BB, 256, 0, stream>>>(lp, targets, in_len, tg_len, num);
  den_reduce_kernel<<<BB, 256, 0, stream>>>(frame_lse, den);
  finalize_kernel<<<1, 32, 0, stream>>>(num, den, out);
}